// Network_Search_Triple_28673201668333
// MI455X (gfx1250) — compile-verified
//
#include <hip/hip_runtime.h>
#include <cstddef>

typedef __attribute__((ext_vector_type(2))) float v2f;
typedef __attribute__((ext_vector_type(8))) float v8f;

#define NFEAT 55
#define DD 64
#define FCW 192
#define REG_C 0.01f

// ---------- helpers ----------

__device__ __forceinline__ float fast_tanh(float x) {
  // tanh(x) = (e^{2x}-1)/(e^{2x}+1), e^{2x} = 2^{2x*log2(e)}
  float t = fminf(fmaxf(x, -9.0f), 9.0f);
  float e = __builtin_amdgcn_exp2f(t * 2.8853900817779268f);
  return (e - 1.0f) * __builtin_amdgcn_rcpf(e + 1.0f);
}

__device__ __forceinline__ float ap(int o, float x, float y) {
  if (o == 0) return x + y;
  if (o == 1) return x * y;
  if (o == 2) return fmaxf(x, y);
  return fminf(x, y);
}

__device__ __forceinline__ float mlp8(float x, const float* __restrict__ w1,
                                      const float* __restrict__ b1,
                                      const float* __restrict__ w2, float b2v) {
  float y = b2v;
#pragma unroll
  for (int j = 0; j < 8; ++j)
    y = fmaf(w2[j], fast_tanh(fmaf(x, w1[j], b1[j])), y);
  return y;
}

// ---------- kernel 1: build effective weights U[55][64] ----------
// Feature order: 0..2 raw p,q,r | 3..14 pair(idx,a) | 15..50 triple(idx,a,bslot)
// | 51..54 extras (plus_plus, multiply_multiply, max_max, min_min).
// prim(idx,a,bslot) = idx*16 + a*4 + bslot; bslot 3 == concat.

__global__ void prep_u(const float* __restrict__ fc, const float* __restrict__ aw,
                       float* __restrict__ U) {
  int i = blockIdx.x * blockDim.x + threadIdx.x;
  if (i >= NFEAT * DD) return;
  int f = i / DD, d = i % DD;
  float u = 0.f;
  if (f == 0) {            // raw p: second operand of '2_a_concat' + concat_concat[0:64]
    for (int a = 0; a < 4; ++a) { int p = 32 + a*4 + 3; u += aw[p] * fc[p*FCW + 64 + d]; }
    u += aw[52] * fc[52*FCW + d];
  } else if (f == 1) {     // raw q: second operand of '1_a_concat' + concat_concat[64:128]
    for (int a = 0; a < 4; ++a) { int p = 16 + a*4 + 3; u += aw[p] * fc[p*FCW + 64 + d]; }
    u += aw[52] * fc[52*FCW + 64 + d];
  } else if (f == 2) {     // raw r: second operand of '0_a_concat' + concat_concat[128:192]
    for (int a = 0; a < 4; ++a) { int p = a*4 + 3;      u += aw[p] * fc[p*FCW + 64 + d]; }
    u += aw[52] * fc[52*FCW + 128 + d];
  } else if (f < 15) {     // pair(idx,a): first operand of 'idx_a_concat'
    int t = f - 3, idx = t / 4, a = t % 4, p = idx*16 + a*4 + 3;
    u = aw[p] * fc[p*FCW + d];
  } else if (f < 51) {     // triple(idx,a,bslot)
    int t = f - 15, idx = t / 12, rem = t % 12, a = rem / 3, bs = rem % 3;
    int p = idx*16 + a*4 + bs;
    u = aw[p] * fc[p*FCW + d];
  } else {                 // extras, prims 48..51
    int p = 48 + (f - 51);
    u = aw[p] * fc[p*FCW + d];
  }
  U[i] = u;
}

// ---------- kernel 2: main fused gather + MLP + WMMA-reduced prim dot ----------

__global__ __launch_bounds__(256) void netsearch_main(
    const int* __restrict__ ps, const int* __restrict__ qs, const int* __restrict__ rs,
    const float* __restrict__ Ep, const float* __restrict__ Eq, const float* __restrict__ Er,
    const float* __restrict__ U,
    const float* __restrict__ pw1, const float* __restrict__ pb1,
    const float* __restrict__ pw2, const float* __restrict__ pb2,
    const float* __restrict__ qw1, const float* __restrict__ qb1,
    const float* __restrict__ qw2, const float* __restrict__ qb2,
    const float* __restrict__ rw1, const float* __restrict__ rb1,
    const float* __restrict__ rw2, const float* __restrict__ rb2,
    float* __restrict__ out, float* __restrict__ blocksums) {
  __shared__ float lu[NFEAT * DD];
  __shared__ float red[8][3];

  for (int i = threadIdx.x; i < NFEAT * DD; i += 256) lu[i] = U[i];
  __syncthreads();

  const int lane  = threadIdx.x & 31;
  const int wave  = threadIdx.x >> 5;
  const int m     = lane & 15;           // row within the 16-row tile (A-matrix M)
  const int khalf = (lane >> 4) << 1;    // 0 for lanes 0-15, 2 for lanes 16-31 (A-matrix K half)
  const int rowbase = (blockIdx.x * 8 + wave) * 16;
  const int row   = rowbase + m;

  const float* prow = Ep + (size_t)ps[row] * DD;
  const float* qrow = Eq + (size_t)qs[row] * DD;
  const float* rrow = Er + (size_t)rs[row] * DD;

  const float pb2s = pb2[0], qb2s = qb2[0], rb2s = rb2[0];

  v8f c = {};
  float sp = 0.f, sq = 0.f, sr = 0.f;

  constexpr int BN[4][3] = {{1,2,3},{0,2,3},{0,1,3},{0,1,2}};  // non-concat b per a

#pragma unroll 1
  for (int dbase = 0; dbase < DD; dbase += 4) {
    const int d0 = dbase + khalf;                 // this lane covers d0, d0+1
    const float2 P = *(const float2*)(prow + d0);
    const float2 Q = *(const float2*)(qrow + d0);
    const float2 R = *(const float2*)(rrow + d0);
    sp = fmaf(P.x, P.x, fmaf(P.y, P.y, sp));
    sq = fmaf(Q.x, Q.x, fmaf(Q.y, Q.y, sq));
    sr = fmaf(R.x, R.x, fmaf(R.y, R.y, sr));

    const float p0 = mlp8(P.x, pw1, pb1, pw2, pb2s);
    const float p1 = mlp8(P.y, pw1, pb1, pw2, pb2s);
    const float q0 = mlp8(Q.x, qw1, qb1, qw2, qb2s);
    const float q1 = mlp8(Q.y, qw1, qb1, qw2, qb2s);
    const float r0 = mlp8(R.x, rw1, rb1, rw2, rb2s);
    const float r1 = mlp8(R.y, rw1, rb1, rw2, rb2s);

    float pq0[4], pq1[4], pr0[4], pr1[4], qr0[4], qr1[4];
#pragma unroll
    for (int a = 0; a < 4; ++a) {
      pq0[a] = ap(a, p0, q0); pq1[a] = ap(a, p1, q1);
      pr0[a] = ap(a, p0, r0); pr1[a] = ap(a, p1, r1);
      qr0[a] = ap(a, q0, r0); qr1[a] = ap(a, q1, r1);
    }

    const float* ub = lu + d0;
    auto wacc = [&](int F, float vx, float vy) {
      v2f A; A.x = vx; A.y = vy;                 // A[m, d0], A[m, d0+1]
      v2f Bm = *(const v2f*)(ub + F * DD);       // B[k,n] = U[F][same d indices], bcast over n
      c = __builtin_amdgcn_wmma_f32_16x16x4_f32(false, A, false, Bm, (short)0, c,
                                                false, false);
    };

    wacc(0, p0, p1);
    wacc(1, q0, q1);
    wacc(2, r0, r1);
#pragma unroll
    for (int a = 0; a < 4; ++a) wacc(3 + a,  pq0[a], pq1[a]);   // pair idx=0 (p,q)
#pragma unroll
    for (int a = 0; a < 4; ++a) wacc(7 + a,  pr0[a], pr1[a]);   // pair idx=1 (p,r)
#pragma unroll
    for (int a = 0; a < 4; ++a) wacc(11 + a, qr0[a], qr1[a]);   // pair idx=2 (q,r)
#pragma unroll
    for (int a = 0; a < 4; ++a) {
#pragma unroll
      for (int bs = 0; bs < 3; ++bs) {
        const int b = BN[a][bs];
        wacc(15 +      a*3 + bs, ap(b, pq0[a], r0), ap(b, pq1[a], r1));  // idx=0
        wacc(15 + 12 + a*3 + bs, ap(b, pr0[a], q0), ap(b, pr1[a], q1));  // idx=1
        wacc(15 + 24 + a*3 + bs, ap(b, qr0[a], p0), ap(b, qr1[a], p1));  // idx=2
      }
    }
    wacc(51, pq0[0] + r0,       pq1[0] + r1);        // plus_plus
    wacc(52, pq0[1] * r0,       pq1[1] * r1);        // multiply_multiply
    wacc(53, fmaxf(pq0[2], r0), fmaxf(pq1[2], r1));  // max_max
    wacc(54, fminf(pq0[3], r0), fminf(pq1[3], r1));  // min_min
  }

  // D layout: lane 0 (N=0) holds M=0..7 in c[0..7]; lane 16 holds M=8..15.
  if (lane == 0) {
#pragma unroll
    for (int j = 0; j < 8; ++j) out[rowbase + j] = c[j];
  }
  if (lane == 16) {
#pragma unroll
    for (int j = 0; j < 8; ++j) out[rowbase + 8 + j] = c[j];
  }

  // Deterministic norm partials: wave shfl tree -> LDS -> per-block sum.
#pragma unroll
  for (int off = 16; off > 0; off >>= 1) {
    sp += __shfl_xor(sp, off, 32);
    sq += __shfl_xor(sq, off, 32);
    sr += __shfl_xor(sr, off, 32);
  }
  if (lane == 0) { red[wave][0] = sp; red[wave][1] = sq; red[wave][2] = sr; }
  __syncthreads();
  if (threadIdx.x == 0) {
    float a0 = 0.f, a1 = 0.f, a2 = 0.f;
#pragma unroll
    for (int w = 0; w < 8; ++w) { a0 += red[w][0]; a1 += red[w][1]; a2 += red[w][2]; }
    blocksums[blockIdx.x * 3 + 0] = a0;
    blocksums[blockIdx.x * 3 + 1] = a1;
    blocksums[blockIdx.x * 3 + 2] = a2;
  }
}

// ---------- kernel 3: deterministic final reduction of the regularizer ----------

__global__ __launch_bounds__(256) void finalize_reg(const float* __restrict__ bs,
                                                    int nblocks, float* __restrict__ out,
                                                    int B) {
  __shared__ float a0[256], a1[256], a2[256];
  float s0 = 0.f, s1 = 0.f, s2 = 0.f;
  for (int i = threadIdx.x; i < nblocks; i += 256) {
    s0 += bs[i * 3 + 0]; s1 += bs[i * 3 + 1]; s2 += bs[i * 3 + 2];
  }
  a0[threadIdx.x] = s0; a1[threadIdx.x] = s1; a2[threadIdx.x] = s2;
  __syncthreads();
  for (int off = 128; off > 0; off >>= 1) {
    if (threadIdx.x < off) {
      a0[threadIdx.x] += a0[threadIdx.x + off];
      a1[threadIdx.x] += a1[threadIdx.x + off];
      a2[threadIdx.x] += a2[threadIdx.x + off];
    }
    __syncthreads();
  }
  if (threadIdx.x == 0)
    out[B] = REG_C * (sqrtf(a0[0]) + sqrtf(a1[0]) + sqrtf(a2[0]));
}

// ---------- launch ----------

extern "C" void kernel_launch(void* const* d_in, const int* in_sizes, int n_in,
                              void* d_out, int out_size, void* d_ws, size_t ws_size,
                              hipStream_t stream) {
  (void)n_in; (void)out_size; (void)ws_size;
  const int*   ps  = (const int*)d_in[0];
  const int*   qs  = (const int*)d_in[1];
  const int*   rs  = (const int*)d_in[2];
  const float* Ep  = (const float*)d_in[3];
  const float* Eq  = (const float*)d_in[4];
  const float* Er  = (const float*)d_in[5];
  const float* fc  = (const float*)d_in[6];
  const float* aw  = (const float*)d_in[7];
  const float* pw1 = (const float*)d_in[8];
  const float* pb1 = (const float*)d_in[9];
  const float* pw2 = (const float*)d_in[10];
  const float* pb2 = (const float*)d_in[11];
  const float* qw1 = (const float*)d_in[12];
  const float* qb1 = (const float*)d_in[13];
  const float* qw2 = (const float*)d_in[14];
  const float* qb2 = (const float*)d_in[15];
  const float* rw1 = (const float*)d_in[16];
  const float* rb1 = (const float*)d_in[17];
  const float* rw2 = (const float*)d_in[18];
  const float* rb2 = (const float*)d_in[19];

  float* out   = (float*)d_out;
  float* U     = (float*)d_ws;                 // NFEAT*DD floats
  float* bsums = U + NFEAT * DD;               // nblocks*3 floats

  const int B = in_sizes[0];                   // 131072
  const int nblocks = B / 128;                 // 16 rows/wave * 8 waves/block

  prep_u<<<(NFEAT * DD + 255) / 256, 256, 0, stream>>>(fc, aw, U);
  netsearch_main<<<nblocks, 256, 0, stream>>>(ps, qs, rs, Ep, Eq, Er, U,
                                              pw1, pb1, pw2, pb2,
                                              qw1, qb1, qw2, qb2,
                                              rw1, rb1, rw2, rb2,
                                              out, bsums);
  finalize_reg<<<1, 256, 0, stream>>>(bsums, nblocks, out, B);
}